// MedianBlur_26353919328316
// MI455X (gfx1250) — compile-verified
//
#include <hip/hip_runtime.h>

// ---------------------------------------------------------------------------
// MedianBlur 3x3, zero-padded, [16,3,512,512] f32.
// HBM-bound: ~100 MB traffic -> ~4.3 us floor @ 23.3 TB/s.
// Staging: CDNA5 async global->LDS (ASYNCcnt):
//   - interior: global_load_async_to_lds_b128 (512 B / wave-instruction,
//     16B-aligned global and LDS addresses)
//   - 2 halo columns: global_load_async_to_lds_b32
//   - zero-padding rows/cols: ds_store
// Compute: 3x3 median min/max network from LDS, 4 px/lane, b128 store.
// ---------------------------------------------------------------------------

#define IMG_H 512
#define IMG_W 512
#define TILE_W 128
#define TILE_H 8
#define HALO_H (TILE_H + 2)      // 10 rows
#define LDS_PITCH 136            // floats; row stride 544 B (mult of 16)
#define CI_BASE 4                // tile col index of gx == x0 (16B aligned)
#define NTHREADS 256

typedef int v4i __attribute__((ext_vector_type(4)));
typedef __attribute__((address_space(1))) int* as1_int_ptr;
typedef __attribute__((address_space(3))) int* as3_int_ptr;
typedef __attribute__((address_space(1))) v4i* as1_v4i_ptr;
typedef __attribute__((address_space(3))) v4i* as3_v4i_ptr;

__device__ __forceinline__ void async_ld_b32(const float* g, float* l) {
#if __has_builtin(__builtin_amdgcn_global_load_async_to_lds_b32)
  __builtin_amdgcn_global_load_async_to_lds_b32(
      (as1_int_ptr)g, (as3_int_ptr)l, /*offset=*/0, /*cpol=*/0);
#else
  *l = *g;
#endif
}

__device__ __forceinline__ void async_ld_b128(const float* g, float* l) {
#if __has_builtin(__builtin_amdgcn_global_load_async_to_lds_b128)
  __builtin_amdgcn_global_load_async_to_lds_b128(
      (as1_v4i_ptr)g, (as3_v4i_ptr)l, /*offset=*/0, /*cpol=*/0);
#else
  *(float4*)l = *(const float4*)g;
#endif
}

__device__ __forceinline__ void wait_async0() {
#if __has_builtin(__builtin_amdgcn_s_wait_asynccnt)
  __builtin_amdgcn_s_wait_asynccnt(0);
#else
  asm volatile("s_wait_asynccnt 0" ::: "memory");
#endif
}

// ---- min/max network helpers ----------------------------------------------
__device__ __forceinline__ float med3(float a, float b, float c) {
  return fmaxf(fminf(a, b), fminf(fmaxf(a, b), c));
}

__device__ __forceinline__ void sort3(float& a, float& b, float& c) {
  float t;
  t = fminf(a, b); b = fmaxf(a, b); a = t;  // a<=b
  t = fminf(b, c); c = fmaxf(b, c); b = t;  // c = max
  t = fminf(a, b); b = fmaxf(a, b); a = t;  // a = min, b = mid
}

__global__ __launch_bounds__(NTHREADS)
void median3x3_kernel(const float* __restrict__ in, float* __restrict__ out) {
  __shared__ float tile[HALO_H * LDS_PITCH];

  const int t  = threadIdx.x;
  const int x0 = blockIdx.x * TILE_W;     // multiple of 128
  const int y0 = blockIdx.y * TILE_H;
  const size_t plane = (size_t)IMG_H * IMG_W;
  const float* __restrict__ inP  = in  + plane * blockIdx.z;
  float* __restrict__       outP = out + plane * blockIdx.z;

  // ---- Interior: 10 rows x 32 b128-slots = 320 slots, async b128 ---------
#pragma unroll
  for (int s = t; s < HALO_H * 32; s += NTHREADS) {
    const int ly = s >> 5;                // 0..9
    const int L  = s & 31;                // b128 slot within row
    const int gy = y0 - 1 + ly;
    float* lp = &tile[ly * LDS_PITCH + CI_BASE + 4 * L];  // 16B aligned
    if ((unsigned)gy < (unsigned)IMG_H) {
      async_ld_b128(inP + (size_t)gy * IMG_W + x0 + 4 * L, lp);
    } else {
      *(float4*)lp = make_float4(0.f, 0.f, 0.f, 0.f);     // top/bottom pad
    }
  }

  // ---- Halo columns: 10 rows x 2 sides = 20 elements, async b32 ----------
  if (t < HALO_H * 2) {
    const int ly   = t >> 1;
    const int side = t & 1;               // 0 = left (gx = x0-1), 1 = right
    const int gy = y0 - 1 + ly;
    const int gx = side ? (x0 + TILE_W) : (x0 - 1);
    const int ci = side ? (CI_BASE + TILE_W) : (CI_BASE - 1);
    float* lp = &tile[ly * LDS_PITCH + ci];
    if ((unsigned)gy < (unsigned)IMG_H && (unsigned)gx < (unsigned)IMG_W) {
      async_ld_b32(inP + (size_t)gy * IMG_W + gx, lp);
    } else {
      *lp = 0.0f;                         // corner / edge pad
    }
  }

  wait_async0();     // drain this wave's ASYNCcnt before barrier
  __syncthreads();   // all waves' LDS writes visible

  // ---- Each lane computes 4 consecutive output pixels --------------------
  const int col = (t & 31) * 4;           // first output x (tile-relative)
  const int row = t >> 5;                 // output y (tile-relative)
  const int base = CI_BASE - 1 + col;     // tile col of window's left edge

  float r0[6], r1[6], r2[6];
#pragma unroll
  for (int k = 0; k < 6; ++k) {
    r0[k] = tile[(row + 0) * LDS_PITCH + base + k];
    r1[k] = tile[(row + 1) * LDS_PITCH + base + k];
    r2[k] = tile[(row + 2) * LDS_PITCH + base + k];
  }

  // Sort the 6 vertical column triples once; shared by the 4 windows.
  float cmin[6], cmid[6], cmax[6];
#pragma unroll
  for (int k = 0; k < 6; ++k) {
    float a = r0[k], b = r1[k], c = r2[k];
    sort3(a, b, c);
    cmin[k] = a; cmid[k] = b; cmax[k] = c;
  }

  float4 o;
  float* op = &o.x;
#pragma unroll
  for (int j = 0; j < 4; ++j) {
    const float lo = fmaxf(fmaxf(cmin[j], cmin[j + 1]), cmin[j + 2]);
    const float mi = med3(cmid[j], cmid[j + 1], cmid[j + 2]);
    const float hi = fminf(fminf(cmax[j], cmax[j + 1]), cmax[j + 2]);
    op[j] = med3(lo, mi, hi);
  }

  // Aligned 16B store (global_store_b128).
  *(float4*)(outP + (size_t)(y0 + row) * IMG_W + (x0 + col)) = o;
}

extern "C" void kernel_launch(void* const* d_in, const int* in_sizes, int n_in,
                              void* d_out, int out_size, void* d_ws, size_t ws_size,
                              hipStream_t stream) {
  const float* in = (const float*)d_in[0];
  float* out = (float*)d_out;
  const int planes = out_size / (IMG_H * IMG_W);  // 16*3 = 48
  dim3 grid(IMG_W / TILE_W, IMG_H / TILE_H, planes);
  median3x3_kernel<<<grid, NTHREADS, 0, stream>>>(in, out);
}